// ModelNew_58798102282512
// MI455X (gfx1250) — compile-verified
//
#include <hip/hip_runtime.h>

typedef __attribute__((ext_vector_type(2))) float v2f;
typedef __attribute__((ext_vector_type(8))) float v8f;

#define KDIM 4096
#define MROWS 65536
#define NCH 16   // partial-sum chunks for the weight reduction

// ---------------------------------------------------------------------------
// Kernel A: partial column sums of weight [N=4096, K=4096] along N.
// Block (bx,by): columns bx*256..bx*256+255, rows by*256..by*256+255.
// Each inner iteration reads 256 consecutive floats of one row -> coalesced.
// ---------------------------------------------------------------------------
__global__ void __launch_bounds__(256)
wsum_partial(const float* __restrict__ w, float* __restrict__ part) {
    const int k  = blockIdx.x * 256 + threadIdx.x;
    const int n0 = blockIdx.y * (KDIM / NCH);
    const float* p = w + (long)n0 * KDIM + k;
    float s = 0.f;
#pragma unroll 8
    for (int n = 0; n < KDIM / NCH; ++n) {
        s += p[(long)n * KDIM];
    }
    part[blockIdx.y * KDIM + k] = s;
}

// ---------------------------------------------------------------------------
// Kernel B: finalize w_red[k] = sum of NCH partials; block 0 also reduces
// bias[4096] -> scalar stored at wfin[KDIM].
// ---------------------------------------------------------------------------
__global__ void __launch_bounds__(256)
wsum_final(const float* __restrict__ part, const float* __restrict__ bias,
           float* __restrict__ wfin) {
    const int k = blockIdx.x * 256 + threadIdx.x;
    float s = 0.f;
#pragma unroll
    for (int c = 0; c < NCH; ++c) s += part[c * KDIM + k];
    wfin[k] = s;

    if (blockIdx.x == 0) {
        __shared__ float red[256];
        float t = 0.f;
        for (int i = threadIdx.x; i < KDIM; i += 256) t += bias[i];
        red[threadIdx.x] = t;
        __syncthreads();
        for (int off = 128; off > 0; off >>= 1) {
            if (threadIdx.x < off) red[threadIdx.x] += red[threadIdx.x + off];
            __syncthreads();
        }
        if (threadIdx.x == 0) wfin[KDIM] = red[0];
    }
}

// ---------------------------------------------------------------------------
// Kernel C: out[m] = x[m,:] . w_red + b_red via V_WMMA_F32_16X16X4_F32.
// 256 threads = 8 waves; each wave owns 16 rows of x (block covers 128 rows).
// A (16x4 f32, documented layout): lane<16 supplies K0,K1; lane>=16 K2,K3.
// B (4x16 f32): w_red[ka], w_red[ka+1] broadcast to all 16 columns,
// same half-wave K-split as A. Every column of D holds the dot product.
// f32 accumulation over K=4096 in 1024 WMMAs per wave.
// ---------------------------------------------------------------------------
__global__ void __launch_bounds__(256)
matvec_wmma(const float* __restrict__ x, const float* __restrict__ wfin,
            float* __restrict__ out) {
    __shared__ float lw[KDIM];
    const int tid = threadIdx.x;
#pragma unroll
    for (int i = 0; i < KDIM / 256; ++i)
        lw[tid + i * 256] = wfin[tid + i * 256];
    __syncthreads();
    const float bsum = wfin[KDIM];

    const int wave = tid >> 5;
    const int lane = tid & 31;
    const int half = lane >> 4;   // 0: lanes 0-15, 1: lanes 16-31
    const int l16  = lane & 15;
    const long m0  = (long)blockIdx.x * 128 + (long)wave * 16;
    const float* xrow = x + (m0 + l16) * (long)KDIM;

    v8f acc = {};
#pragma unroll 8
    for (int kk = 0; kk < KDIM; kk += 4) {
        const int ka = kk + half * 2;
        v2f a, b;
        a.x = xrow[ka];
        a.y = xrow[ka + 1];
        b.x = lw[ka];
        b.y = lw[ka + 1];
        // (neg_a, A, neg_b, B, c_mod, C, reuse_a, reuse_b)
        acc = __builtin_amdgcn_wmma_f32_16x16x4_f32(
            false, a, false, b, (short)0, acc, false, false);
    }

    // D layout: VGPR j holds M=j (lanes 0-15) / M=j+8 (lanes 16-31), all
    // columns identical. Lanes 0-7 write rows m0+0..7, lanes 16-23 rows +8..15.
    if (l16 < 8) {
        out[m0 + half * 8 + l16] = acc[l16] + bsum;
    }
}

// ---------------------------------------------------------------------------
extern "C" void kernel_launch(void* const* d_in, const int* in_sizes, int n_in,
                              void* d_out, int out_size, void* d_ws, size_t ws_size,
                              hipStream_t stream) {
    const float* x    = (const float*)d_in[0];
    const float* wgt  = (const float*)d_in[1];
    const float* bias = (const float*)d_in[2];
    float* out  = (float*)d_out;
    float* part = (float*)d_ws;            // NCH*KDIM floats
    float* wfin = part + NCH * KDIM;       // KDIM+1 floats (w_red + b_red)

    wsum_partial<<<dim3(KDIM / 256, NCH), 256, 0, stream>>>(wgt, part);
    wsum_final  <<<dim3(KDIM / 256), 256, 0, stream>>>(part, bias, wfin);
    matvec_wmma <<<dim3(MROWS / 128), 256, 0, stream>>>(x, wfin, out);
}